// SplineDepthHeadV2_25615184954140
// MI455X (gfx1250) — compile-verified
//
#include <hip/hip_runtime.h>
#include <hip/hip_bf16.h>

// ---- problem constants (match reference) ----
#define BATCH   128
#define LATENT  512
#define RANK    16
#define HD      256
#define NHEADS  4
#define OH      512
#define OW      512
#define CXY     16

typedef __attribute__((ext_vector_type(2))) float v2f;
typedef __attribute__((ext_vector_type(8))) float v8f;

// D(16x16,f32) = A(16x4,f32) x B(4x16,f32) + C  -- native f32 WMMA on gfx1250
__device__ __forceinline__ v8f wmma4(v2f a, v2f b, v8f c) {
  return __builtin_amdgcn_wmma_f32_16x16x4_f32(false, a, false, b, (short)0, c,
                                               false, false);
}

// =====================================================================
// K1: f[b,r,:] = leaky( [h[b] | rank_emb[r]] @ rW1[r] + rb1[r], 0.2 )
// grid (B/16, HD/16, RANK), block 32 (one wave -> one 16x16 tile)
// =====================================================================
__global__ __launch_bounds__(32) void k_rank_l1(
    const float* __restrict__ h, const float* __restrict__ remb,
    const float* __restrict__ rW1, const float* __restrict__ rb1,
    float* __restrict__ f) {
  const int r    = blockIdx.z;
  const int lane = threadIdx.x & 31;
  const int hv   = lane >> 4;      // half-wave: selects K pair {0,1} vs {2,3}
  const int lo   = lane & 15;
  const int row  = blockIdx.x * 16 + lo;   // batch index (A rows)
  const int col  = blockIdx.y * 16 + lo;   // output channel (B cols)
  const float* Bw = rW1 + (long long)r * 768 * HD;

  v8f acc = {};
  for (int k0 = 0; k0 < LATENT + HD; k0 += 4) {
    const int kb = k0 + 2 * hv;
    v2f a, bv;
    const float* ap = (kb < LATENT) ? (h + (long long)row * LATENT + kb)
                                    : (remb + r * HD + (kb - LATENT));
    a.x  = ap[0];
    a.y  = ap[1];
    bv.x = Bw[kb * HD + col];
    bv.y = Bw[(kb + 1) * HD + col];
    acc = wmma4(a, bv, acc);
  }
#pragma unroll
  for (int j = 0; j < 8; ++j) {
    const int b  = blockIdx.x * 16 + j + 8 * hv;
    float v = acc[j] + rb1[r * HD + col];
    v = v > 0.f ? v : 0.2f * v;
    f[((long long)b * RANK + r) * HD + col] = v;
  }
}

// =====================================================================
// Generic rank-strided GEMM: C = act(A @ B + bias)
// A row-major (lda), per-rank base offset aRS; likewise B/bias/C.
// grid (Mtiles, Ntiles, ranks), block 32.
// =====================================================================
__global__ __launch_bounds__(32) void k_gemm(
    const float* __restrict__ A, long long aRS, int lda,
    const float* __restrict__ Bm, long long bRS, int ldb,
    const float* __restrict__ bias, long long biasRS,
    float* __restrict__ C, long long cRS, int ldc, int K, int act) {
  const int r = blockIdx.z;
  const float* Ar = A + (long long)r * aRS;
  const float* Br = Bm + (long long)r * bRS;
  const float* br = bias + (long long)r * biasRS;
  float* Cr = C + (long long)r * cRS;

  const int lane = threadIdx.x & 31;
  const int hv   = lane >> 4;
  const int lo   = lane & 15;
  const int row  = blockIdx.x * 16 + lo;
  const int col  = blockIdx.y * 16 + lo;

  v8f acc = {};
  for (int k0 = 0; k0 < K; k0 += 4) {
    const int kb = k0 + 2 * hv;
    v2f a, bv;
    a.x  = Ar[(long long)row * lda + kb];
    a.y  = Ar[(long long)row * lda + kb + 1];
    bv.x = Br[(long long)kb * ldb + col];
    bv.y = Br[(long long)(kb + 1) * ldb + col];
    acc = wmma4(a, bv, acc);
  }
#pragma unroll
  for (int j = 0; j < 8; ++j) {
    const int m = blockIdx.x * 16 + j + 8 * hv;
    float v = acc[j] + br[col];
    if (act) v = v > 0.f ? v : 0.2f * v;
    Cr[(long long)m * ldc + col] = v;
  }
}

// =====================================================================
// K4: attention per (b, head). seq = RANK = 16, head_dim = 64.
// scores via WMMA, shuffle softmax within 16-lane halves, att@V via WMMA.
// grid (B*NHEADS), block 32.
// =====================================================================
__global__ __launch_bounds__(32) void k_attn(const float* __restrict__ qkv,
                                             float* __restrict__ ao) {
  __shared__ float latt[16][16];
  const int b    = blockIdx.x >> 2;
  const int head = blockIdx.x & 3;
  const int lane = threadIdx.x & 31;
  const int hv   = lane >> 4;
  const int lo   = lane & 15;

  const float* qb = qkv + (long long)b * RANK * 768 + head * 64;
  const float* kb = qb + HD;       // K block at +256
  const float* vb = qb + 2 * HD;   // V block at +512

  // scores = Q @ K^T : B[kk][n] = Kmat[n][kk] (same gather pattern as A)
  v8f s = {};
#pragma unroll
  for (int k0 = 0; k0 < 64; k0 += 4) {
    const int kk = k0 + 2 * hv;
    v2f a, bv;
    a.x  = qb[lo * 768 + kk];
    a.y  = qb[lo * 768 + kk + 1];
    bv.x = kb[lo * 768 + kk];
    bv.y = kb[lo * 768 + kk + 1];
    s = wmma4(a, bv, s);
  }

  // softmax over each 16-wide row (rows live across 16-lane halves)
  float att[8];
#pragma unroll
  for (int j = 0; j < 8; ++j) {
    float val = s[j] * 0.125f;  // 1/sqrt(64)
    float m = val;
    for (int o = 1; o < 16; o <<= 1) m = fmaxf(m, __shfl_xor(m, o, 32));
    const float e = expf(val - m);
    float sum = e;
    for (int o = 1; o < 16; o <<= 1) sum += __shfl_xor(sum, o, 32);
    att[j] = e / sum;
  }
#pragma unroll
  for (int j = 0; j < 8; ++j) latt[j + 8 * hv][lo] = att[j];
  __syncthreads();

  // O = att(16x16) @ V(16x64)
#pragma unroll
  for (int n0 = 0; n0 < 64; n0 += 16) {
    v8f o8 = {};
#pragma unroll
    for (int k0 = 0; k0 < 16; k0 += 4) {
      const int kk = k0 + 2 * hv;
      v2f a, bv;
      a.x  = latt[lo][kk];
      a.y  = latt[lo][kk + 1];
      bv.x = vb[kk * 768 + n0 + lo];
      bv.y = vb[(kk + 1) * 768 + n0 + lo];
      o8 = wmma4(a, bv, o8);
    }
#pragma unroll
    for (int j = 0; j < 8; ++j) {
      const int rrow = j + 8 * hv;
      ao[((long long)b * RANK + rrow) * HD + head * 64 + n0 + lo] = o8[j];
    }
  }
}

// =====================================================================
// K6b: control head.  out = (rf@gW + gb) * (fxy@cW2 + cb2) + (rf@bW + bb)
// Three accumulators share one K-loop. grid (B/16, RANK), block 32.
// =====================================================================
__global__ __launch_bounds__(32) void k_ctrl2(
    const float* __restrict__ rf, const float* __restrict__ fxy,
    const float* __restrict__ cW2, const float* __restrict__ cb2,
    const float* __restrict__ gW, const float* __restrict__ gb,
    const float* __restrict__ bW, const float* __restrict__ bb,
    float* __restrict__ outp) {
  const int r    = blockIdx.y;
  const int lane = threadIdx.x & 31;
  const int hv   = lane >> 4;
  const int lo   = lane & 15;
  const int row  = blockIdx.x * 16 + lo;   // batch
  const int col  = lo;                     // channel 0..15

  const float* rfr = rf + r * HD;              // lda = RANK*HD
  const float* fr  = fxy + r * HD;
  const float* W2  = cW2 + (long long)r * HD * CXY;

  v8f pacc = {}, gacc = {}, bacc = {};
  for (int k0 = 0; k0 < HD; k0 += 4) {
    const int kb = k0 + 2 * hv;
    v2f af, ar, b1, b2, b3;
    af.x = fr[(long long)row * (RANK * HD) + kb];
    af.y = fr[(long long)row * (RANK * HD) + kb + 1];
    ar.x = rfr[(long long)row * (RANK * HD) + kb];
    ar.y = rfr[(long long)row * (RANK * HD) + kb + 1];
    b1.x = W2[kb * CXY + col];  b1.y = W2[(kb + 1) * CXY + col];
    b2.x = gW[kb * CXY + col];  b2.y = gW[(kb + 1) * CXY + col];
    b3.x = bW[kb * CXY + col];  b3.y = bW[(kb + 1) * CXY + col];
    pacc = wmma4(af, b1, pacc);
    gacc = wmma4(ar, b2, gacc);
    bacc = wmma4(ar, b3, bacc);
  }
#pragma unroll
  for (int j = 0; j < 8; ++j) {
    const int b  = blockIdx.x * 16 + j + 8 * hv;
    const float p  = pacc[j] + cb2[r * CXY + col];
    const float g  = gacc[j] + gb[col];
    const float bi = bacc[j] + bb[col];
    outp[((long long)b * RANK + r) * CXY + col] = g * p + bi;
  }
}

// =====================================================================
// K7: cubic Hermite spline eval. z=0: u from px; z=1: v from py, folded
// with softmax(rank_weights)[r]. One thread per output sample.
// =====================================================================
__device__ __forceinline__ float spline_tangent(const float* cp, int k) {
  float d;
  if (k == 0)       d = (cp[1]  - cp[0])  * 0.5f;
  else if (k == 15) d = (cp[15] - cp[14]) * 0.5f;
  else              d = (cp[k + 1] - cp[k - 1]) * 0.5f;
  return d * (1.0f / 15.0f);
}

__global__ void k_spline(const float* __restrict__ px,
                         const float* __restrict__ py,
                         const float* __restrict__ rw,
                         float* __restrict__ u, float* __restrict__ v) {
  const int z = blockIdx.z;
  const long long idx = (long long)blockIdx.x * blockDim.x + threadIdx.x;
  const long long total = (long long)BATCH * RANK * OW;
  if (idx >= total) return;

  const int w  = (int)(idx & 511);
  const long long br = idx >> 9;
  const int r  = (int)(br & (RANK - 1));
  const float* cp = (z ? py : px) + br * CXY;

  const float t   = 0.001f + (0.998f / 511.0f) * (float)w;
  const float ts  = t * 15.0f;
  int seg = (int)floorf(ts);
  seg = min(max(seg, 0), 14);
  float tau = ts - (float)seg;
  tau = fminf(fmaxf(tau, 0.0f), 0.9999f);
  const float tau2 = tau * tau, tau3 = tau2 * tau;
  const float h00 = 2.f * tau3 - 3.f * tau2 + 1.f;
  const float h10 = tau3 - 2.f * tau2 + tau;
  const float h01 = -2.f * tau3 + 3.f * tau2;
  const float h11 = tau3 - tau2;

  const float pk  = cp[seg];
  const float pk1 = cp[seg + 1];
  const float mk  = spline_tangent(cp, seg);
  const float mk1 = spline_tangent(cp, seg + 1);
  float res = h00 * pk + h10 * mk + h01 * pk1 + h11 * mk1;

  if (z) {  // fold softmax(rank_weights)[r] into v
    float m = rw[0];
    for (int i = 1; i < RANK; ++i) m = fmaxf(m, rw[i]);
    float sum = 0.f;
    for (int i = 0; i < RANK; ++i) sum += expf(rw[i] - m);
    res *= expf(rw[r] - m) / sum;
    v[idx] = res;
  } else {
    u[idx] = res;
  }
}

// =====================================================================
// K8: depth[b,h,w] = sum_r v[b,r,h]*u[b,r,w] + gbias
// Per-batch GEMM [512x16]x[16x512] on WMMA; 8 waves/block, 1 tile/wave.
// Store-bandwidth bound (134 MB).
// =====================================================================
__global__ __launch_bounds__(256) void k_final(const float* __restrict__ U,
                                               const float* __restrict__ V,
                                               const float* __restrict__ gbias,
                                               float* __restrict__ outp) {
  const int wave = blockIdx.x * 8 + (threadIdx.x >> 5);
  const int lane = threadIdx.x & 31;
  const int hv   = lane >> 4;
  const int lo   = lane & 15;

  const int b   = wave >> 10;        // / (32*32)
  const int rem = wave & 1023;
  const int tH  = rem >> 5;
  const int tW  = rem & 31;

  const float* Vb = V + (long long)b * RANK * OH;
  const float* Ub = U + (long long)b * RANK * OW;

  v8f acc = {};
#pragma unroll
  for (int k0 = 0; k0 < RANK; k0 += 4) {
    const int kb = k0 + 2 * hv;
    v2f a, bv;
    a.x  = Vb[kb * OH + tH * 16 + lo];        // A = V^T (transposed gather)
    a.y  = Vb[(kb + 1) * OH + tH * 16 + lo];
    bv.x = Ub[kb * OW + tW * 16 + lo];
    bv.y = Ub[(kb + 1) * OW + tW * 16 + lo];
    acc = wmma4(a, bv, acc);
  }
  const float g = gbias[0];
#pragma unroll
  for (int j = 0; j < 8; ++j) {
    const int hh = tH * 16 + j + 8 * hv;
    const int ww = tW * 16 + lo;
    outp[(long long)b * OH * OW + (long long)hh * OW + ww] = acc[j] + g;
  }
}

// =====================================================================
// Host launcher
// =====================================================================
extern "C" void kernel_launch(void* const* d_in, const int* in_sizes, int n_in,
                              void* d_out, int out_size, void* d_ws,
                              size_t ws_size, hipStream_t stream) {
  const float* h     = (const float*)d_in[0];
  const float* remb  = (const float*)d_in[1];
  const float* rW1   = (const float*)d_in[2];
  const float* rb1   = (const float*)d_in[3];
  const float* rW2   = (const float*)d_in[4];
  const float* rb2   = (const float*)d_in[5];
  const float* Wqkv  = (const float*)d_in[6];
  const float* bqkv  = (const float*)d_in[7];
  const float* Wo    = (const float*)d_in[8];
  const float* bo    = (const float*)d_in[9];
  const float* Wout  = (const float*)d_in[10];
  const float* bout  = (const float*)d_in[11];
  const float* cxW1  = (const float*)d_in[12];
  const float* cxb1  = (const float*)d_in[13];
  const float* cxW2  = (const float*)d_in[14];
  const float* cxb2  = (const float*)d_in[15];
  const float* cyW1  = (const float*)d_in[16];
  const float* cyb1  = (const float*)d_in[17];
  const float* cyW2  = (const float*)d_in[18];
  const float* cyb2  = (const float*)d_in[19];
  const float* gxW   = (const float*)d_in[20];
  const float* gxb   = (const float*)d_in[21];
  const float* bxW   = (const float*)d_in[22];
  const float* bxb   = (const float*)d_in[23];
  const float* gyW   = (const float*)d_in[24];
  const float* gyb   = (const float*)d_in[25];
  const float* byW   = (const float*)d_in[26];
  const float* byb   = (const float*)d_in[27];
  const float* rw    = (const float*)d_in[28];
  const float* gbias = (const float*)d_in[29];

  float* out = (float*)d_out;
  float* ws  = (float*)d_ws;

  const long long NBR = (long long)BATCH * RANK;           // 2048
  float* f   = ws;                      // [B,R,HD]   524288
  float* x0  = f   + 524288;            // [B,R,HD]
  float* qkv = x0  + 524288;            // [B*R,768]  1572864
  float* ao  = qkv + 1572864;           // [B,R,HD]
  float* x1  = ao  + 524288;            // [B,R,HD]
  float* x2  = x1  + 524288;            // [B,R,HD]  (final rank_feats)
  float* px  = x2  + 524288;            // [B,R,16]
  float* py  = px  + 32768;             // [B,R,16]
  float* u   = py  + 32768;             // [B,R,512]
  float* v   = u   + 1048576;           // [B,R,512]
  float* fx  = f;                       // reuse (f dead after K2)
  float* fy  = x0;                      // reuse (x0 dead after K3)

  // K1: layer-1 per-rank MLP with concat loader
  k_rank_l1<<<dim3(BATCH / 16, HD / 16, RANK), 32, 0, stream>>>(h, remb, rW1,
                                                                rb1, f);
  // K2: rank_feats_pre = f @ rW2[r] + rb2[r]
  k_gemm<<<dim3(BATCH / 16, HD / 16, RANK), 32, 0, stream>>>(
      f, HD, RANK * HD, rW2, (long long)HD * HD, HD, rb2, HD, x0, HD,
      RANK * HD, HD, 0);
  // K3: qkv = x0 @ Wqkv + bqkv   ([2048,256]x[256,768])
  k_gemm<<<dim3((int)(NBR / 16), 48, 1), 32, 0, stream>>>(
      x0, 0, HD, Wqkv, 0, 3 * HD, bqkv, 0, qkv, 0, 3 * HD, HD, 0);
  // K4: attention
  k_attn<<<BATCH * NHEADS, 32, 0, stream>>>(qkv, ao);
  // K5a: x1 = ao @ Wo + bo
  k_gemm<<<dim3((int)(NBR / 16), HD / 16, 1), 32, 0, stream>>>(
      ao, 0, HD, Wo, 0, HD, bo, 0, x1, 0, HD, HD, 0);
  // K5b: rank_feats = x1 @ Wout + bout
  k_gemm<<<dim3((int)(NBR / 16), HD / 16, 1), 32, 0, stream>>>(
      x1, 0, HD, Wout, 0, HD, bout, 0, x2, 0, HD, HD, 0);
  // K6a: fx = leaky(rank_feats @ cxW1[r] + cxb1[r]);  fy likewise
  k_gemm<<<dim3(BATCH / 16, HD / 16, RANK), 32, 0, stream>>>(
      x2, HD, RANK * HD, cxW1, (long long)HD * HD, HD, cxb1, HD, fx, HD,
      RANK * HD, HD, 1);
  k_gemm<<<dim3(BATCH / 16, HD / 16, RANK), 32, 0, stream>>>(
      x2, HD, RANK * HD, cyW1, (long long)HD * HD, HD, cyb1, HD, fy, HD,
      RANK * HD, HD, 1);
  // K6b: px / py with fused gate+bias modulation
  k_ctrl2<<<dim3(BATCH / 16, RANK), 32, 0, stream>>>(x2, fx, cxW2, cxb2, gxW,
                                                     gxb, bxW, bxb, px);
  k_ctrl2<<<dim3(BATCH / 16, RANK), 32, 0, stream>>>(x2, fy, cyW2, cyb2, gyW,
                                                     gyb, byW, byb, py);
  // K7: splines (z=0 -> u, z=1 -> v scaled by softmax(rank_weights))
  {
    const long long total = (long long)BATCH * RANK * OW;
    k_spline<<<dim3((unsigned)((total + 255) / 256), 1, 2), 256, 0, stream>>>(
        px, py, rw, u, v);
  }
  // K8: outer-product expansion as per-batch WMMA GEMM + global_bias
  {
    const int tiles = BATCH * (OH / 16) * (OW / 16);  // 131072
    k_final<<<tiles / 8, 256, 0, stream>>>(u, v, gbias, out);
  }
}